// MultiModalFusion_76605036692279
// MI455X (gfx1250) — compile-verified
//
#include <hip/hip_runtime.h>
#include <math.h>

typedef float v2f __attribute__((ext_vector_type(2)));
typedef float v8f __attribute__((ext_vector_type(8)));

#define NB 2
#define NC 64
#define NVOX 131072   // 8*128*128
#define YXN 16384     // 128*128
#define NSLAB 128
#define KSLAB 1024    // NVOX / NSLAB

// ---- ws layout (floats) ----
#define WS_PARTIAL 0               // 2*128*4096 = 1048576
#define WS_WA      1048576         // 2*4096     = 8192
#define WS_PM      1056768         // 2*131072   = 262144
#define WS_IM      1318912         // 2*131072   = 262144
// total 1581056 floats = ~6.03 MB

__device__ __forceinline__ v8f wmma_f32(v2f a, v2f b, v8f c) {
  // V_WMMA_F32_16X16X4_F32 : D = A(16x4) x B(4x16) + C(16x16), all fp32
  return __builtin_amdgcn_wmma_f32_16x16x4_f32(false, a, false, b, (short)0, c,
                                               false, false);
}

// LDS byte offset of a __shared__ pointer: generic LDS addresses carry the
// LDS offset in the low 32 bits (aperture layout, ISA 10.2).
__device__ __forceinline__ unsigned lds_off(const void* p) {
  return (unsigned)(size_t)p;
}

// Async global->LDS copy of 16 bytes per lane (ASYNCcnt-tracked, CDNA5).
__device__ __forceinline__ void async_cp16(unsigned lds_addr, const float* g) {
  asm volatile("global_load_async_to_lds_b128 %0, %1, off"
               :: "v"(lds_addr), "v"(g)
               : "memory");
}

__device__ __forceinline__ void wait_async0() {
  asm volatile("s_wait_asynccnt 0" ::: "memory");
}

// ---------------------------------------------------------------------------
// Kernel 1: score[b,c,d] = sum_n pc[b,c,n] * img[b,d,n]   (GEMM, K = voxels)
// One WG per (batch, K-slab of 1024 voxels); partial 64x64 per slab to ws.
// Double-buffered async global->LDS staging.
// ---------------------------------------------------------------------------
#define K1_SQ0 0          // 64*68
#define K1_SK0 4352
#define K1_SQ1 8704
#define K1_SK1 13056
#define K1_LDS_FLOATS 17408
#define K1_LDS_BYTES (K1_LDS_FLOATS * 4)

__device__ __forceinline__ void k1_prefetch(const float* pcb, const float* imb,
                                            int kb, float* sQ, float* sK,
                                            int tid) {
#pragma unroll
  for (int j = 0; j < 4; ++j) {
    int lin = j * 256 + tid;            // float4 id, 1024 total
    int row = lin >> 4;
    int c4  = (lin & 15) * 4;
    async_cp16(lds_off(sQ + row * 68 + c4), pcb + (size_t)row * NVOX + kb + c4);
    async_cp16(lds_off(sK + row * 68 + c4), imb + (size_t)row * NVOX + kb + c4);
  }
}

__global__ void score_partials(const float* __restrict__ img,
                               const float* __restrict__ pc,
                               float* __restrict__ partial) {
  extern __shared__ float sm1[];
  float* sQ0 = sm1 + K1_SQ0;
  float* sK0 = sm1 + K1_SK0;
  float* sQ1 = sm1 + K1_SQ1;
  float* sK1 = sm1 + K1_SK1;

  const int b    = blockIdx.x / NSLAB;
  const int s    = blockIdx.x % NSLAB;
  const int tid  = threadIdx.x;
  const int lane = tid & 31;
  const int w    = tid >> 5;       // wave 0..7
  const int hi   = lane >> 4;      // half-wave select (K +2 per ISA layout)
  const int lr   = lane & 15;

  const float* pcb = pc  + (size_t)b * NC * NVOX;
  const float* imb = img + (size_t)b * NC * NVOX;

  const int mt  = w & 3;           // M-tile (pc channel block)
  const int ntb = (w >> 2) * 2;    // two N-tiles (img channel blocks)
  v8f acc0 = {}; v8f acc1 = {};

  k1_prefetch(pcb, imb, s * KSLAB, sQ0, sK0, tid);

  for (int it = 0; it < 16; ++it) {
    float* sQ = (it & 1) ? sQ1 : sQ0;
    float* sK = (it & 1) ? sK1 : sK0;
    wait_async0();                 // current tile's async copies done (per wave)
    __syncthreads();               // visible to all waves
    if (it < 15)                   // overlap next tile copy with WMMA compute
      k1_prefetch(pcb, imb, s * KSLAB + (it + 1) * 64,
                  (it & 1) ? sQ0 : sQ1, (it & 1) ? sK0 : sK1, tid);
#pragma unroll
    for (int kk = 0; kk < 16; ++kk) {
      const int k0 = kk * 4 + hi * 2;
      v2f a;  a.x  = sQ[(mt * 16 + lr) * 68 + k0];
              a.y  = sQ[(mt * 16 + lr) * 68 + k0 + 1];
      v2f b0; b0.x = sK[(ntb * 16 + lr) * 68 + k0];
              b0.y = sK[(ntb * 16 + lr) * 68 + k0 + 1];
      v2f b1; b1.x = sK[((ntb + 1) * 16 + lr) * 68 + k0];
              b1.y = sK[((ntb + 1) * 16 + lr) * 68 + k0 + 1];
      acc0 = wmma_f32(a, b0, acc0);
      acc1 = wmma_f32(a, b1, acc1);
    }
    // no trailing barrier: next iteration's wait+barrier precedes any
    // overwrite of the buffer this iteration read (two-deep rotation)
  }
  float* p = partial + (size_t)(b * NSLAB + s) * 4096;
#pragma unroll
  for (int r = 0; r < 8; ++r) {
    int M = mt * 16 + r + hi * 8;                 // C/D layout: row r (+8 hi)
    p[M * 64 + ntb * 16 + lr]       = acc0[r];
    p[M * 64 + (ntb + 1) * 16 + lr] = acc1[r];
  }
}

// ---------------------------------------------------------------------------
// Kernel 2: reduce partials, softmax rows, fold Wa = Wp @ attn  (per batch)
// ---------------------------------------------------------------------------
__global__ void softmax_wa(const float* __restrict__ partial,
                           const float* __restrict__ Wp,
                           float* __restrict__ wsWa) {
  const int b   = blockIdx.x;
  const int tid = threadIdx.x;
  __shared__ float sc[4096];
#pragma unroll
  for (int e = 0; e < 16; ++e) {
    int idx = tid * 16 + e;
    float s = 0.f;
    for (int sl = 0; sl < NSLAB; ++sl)
      s += partial[(size_t)(b * NSLAB + sl) * 4096 + idx];
    sc[idx] = s * 0.125f;   // 1/sqrt(C), C=64
  }
  __syncthreads();
  if (tid < 64) {           // row-wise softmax over d
    float mx = -1e30f;
    for (int d = 0; d < 64; ++d) mx = fmaxf(mx, sc[tid * 64 + d]);
    float sum = 0.f;
    for (int d = 0; d < 64; ++d) {
      float e = __expf(sc[tid * 64 + d] - mx);
      sc[tid * 64 + d] = e; sum += e;
    }
    float inv = 1.f / sum;
    for (int d = 0; d < 64; ++d) sc[tid * 64 + d] *= inv;
  }
  __syncthreads();
#pragma unroll
  for (int e = 0; e < 16; ++e) {       // Wa[d][k] = sum_c Wp[d][c]*attn[c][k]
    int idx = tid * 16 + e;
    int d = idx >> 6, col = idx & 63;
    float s = 0.f;
    for (int c = 0; c < 64; ++c) s += Wp[d * 64 + c] * sc[c * 64 + col];
    wsWa[b * 4096 + idx] = s;
  }
}

// ---------------------------------------------------------------------------
// Kernel 3: fused main pass, double-buffered async tile staging.
// ---------------------------------------------------------------------------
#define ACT_S 132                 // padded stride for 128-voxel activation rows
#define OFF_WG   0                // 64*68
#define OFF_WF   4352             // 64*132
#define OFF_WA   12800            // 64*68
#define OFF_PC0  17152            // 64*132
#define OFF_IG0  25600            // 64*132
#define OFF_PC1  34048            // 64*132
#define OFF_IG1  42496            // 64*132
#define OFF_G    50944            // 64*132
#define OFF_BG   59392
#define OFF_BF   59456
#define OFF_BP   59520
#define OFF_PM   59584            // 128
#define OFF_IMK  59712            // 128
#define K3_LDS_FLOATS 59840
#define K3_LDS_BYTES (K3_LDS_FLOATS * 4)   // 239,360 B < 320 KB WGP limit

__device__ __forceinline__ void k3_prefetch(const float* pcb, const float* imb,
                                            int n0, float* sPc, float* sIg,
                                            int tid) {
#pragma unroll
  for (int j = 0; j < 8; ++j) {
    int lin = j * 256 + tid;            // float4 id, 2048 total
    int ch  = lin >> 5;
    int col = (lin & 31) * 4;
    async_cp16(lds_off(sPc + ch * ACT_S + col),
               pcb + (size_t)ch * NVOX + n0 + col);
    async_cp16(lds_off(sIg + ch * ACT_S + col),
               imb + (size_t)ch * NVOX + n0 + col);
  }
}

__global__ void fused_main(const float* __restrict__ img,
                           const float* __restrict__ pc,
                           const float* __restrict__ Wg,
                           const float* __restrict__ bg,
                           const float* __restrict__ Wf,
                           const float* __restrict__ bfp,
                           const float* __restrict__ bp,
                           const float* __restrict__ wsWa,
                           float* __restrict__ wsPm,
                           float* __restrict__ wsIm,
                           float* __restrict__ out) {
  extern __shared__ float sm[];
  float* sWg  = sm + OFF_WG;
  float* sWf  = sm + OFF_WF;
  float* sWa  = sm + OFF_WA;
  float* sPc0 = sm + OFF_PC0;
  float* sIg0 = sm + OFF_IG0;
  float* sPc1 = sm + OFF_PC1;
  float* sIg1 = sm + OFF_IG1;
  float* sG   = sm + OFF_G;
  float* sBg  = sm + OFF_BG;
  float* sBf  = sm + OFF_BF;
  float* sBp  = sm + OFF_BP;
  float* sPm  = sm + OFF_PM;
  float* sIm  = sm + OFF_IMK;

  const int b    = blockIdx.x >> 7;
  const int wgi  = blockIdx.x & 127;
  const int tid  = threadIdx.x;
  const int lane = tid & 31;
  const int w    = tid >> 5;
  const int hi   = lane >> 4;
  const int lr   = lane & 15;

  const float* pcb = pc  + (size_t)b * NC * NVOX;
  const float* imb = img + (size_t)b * NC * NVOX;

  // kick off async copy of tile 0 before weight staging
  k3_prefetch(pcb, imb, wgi * 8 * 128, sPc0, sIg0, tid);

  // stage weights (bank-padded) + biases
#pragma unroll
  for (int j = 0; j < 16; ++j) {
    int lin = j * 256 + tid;                      // 4096
    sWg[(lin >> 6) * 68 + (lin & 63)] = Wg[lin];
    sWa[(lin >> 6) * 68 + (lin & 63)] = wsWa[b * 4096 + lin];
  }
#pragma unroll
  for (int j = 0; j < 32; ++j) {
    int lin = j * 256 + tid;                      // 8192
    sWf[(lin >> 7) * 132 + (lin & 127)] = Wf[lin];
  }
  if (tid < 64) { sBg[tid] = bg[tid]; sBf[tid] = bfp[tid]; sBp[tid] = bp[tid]; }

  const v8f vzero = {};

  for (int t = 0; t < 8; ++t) {
    float* sPc = (t & 1) ? sPc1 : sPc0;
    float* sIg = (t & 1) ? sIg1 : sIg0;
    const int n0 = (wgi * 8 + t) * 128;

    wait_async0();                 // this tile's async copies done (per wave)
    __syncthreads();               // all waves' copies + weight stores visible
    if (t < 7)                     // overlap next tile copy with WMMA compute
      k3_prefetch(pcb, imb, n0 + 128,
                  (t & 1) ? sPc0 : sPc1, (t & 1) ? sIg0 : sIg1, tid);

    if (tid < 128) {                              // occupancy masks per voxel
      int v = tid;
      float pm = 0.f, im = 0.f;
      for (int c = 0; c < 64; ++c) {
        if (sPc[c * ACT_S + v] != 0.f) pm = 1.f;
        if (sIg[c * ACT_S + v] != 0.f) im = 1.f;
      }
      sPm[v] = pm; sIm[v] = im;
      wsPm[(size_t)b * NVOX + n0 + v] = pm;
      wsIm[(size_t)b * NVOX + n0 + v] = im;
    }
    __syncthreads();

    const int vcol = w * 16 + lr;                 // this wave's voxel column

    // GEMM1: g = relu(Wg @ pc + bg) * img
    v8f accg[4]; accg[0]=vzero; accg[1]=vzero; accg[2]=vzero; accg[3]=vzero;
#pragma unroll
    for (int kk = 0; kk < 16; ++kk) {
      int k0 = kk * 4 + hi * 2;
      v2f bb; bb.x = sPc[k0 * ACT_S + vcol];
              bb.y = sPc[(k0 + 1) * ACT_S + vcol];
#pragma unroll
      for (int m = 0; m < 4; ++m) {
        v2f aa; aa.x = sWg[(m * 16 + lr) * 68 + k0];
                aa.y = sWg[(m * 16 + lr) * 68 + k0 + 1];
        accg[m] = wmma_f32(aa, bb, accg[m]);
      }
    }
#pragma unroll
    for (int m = 0; m < 4; ++m)
#pragma unroll
      for (int r = 0; r < 8; ++r) {
        int ch = m * 16 + r + hi * 8;
        float g = fmaxf(accg[m][r] + sBg[ch], 0.f) * sIg[ch * ACT_S + vcol];
        sG[ch * ACT_S + vcol] = g;                // own columns only; no barrier
      }

    // GEMM2: f = Wf @ [g ; pc]   (K = 128)
    v8f accf[4]; accf[0]=vzero; accf[1]=vzero; accf[2]=vzero; accf[3]=vzero;
#pragma unroll
    for (int kk = 0; kk < 32; ++kk) {
      int k0 = kk * 4 + hi * 2;
      const float* src = (k0 < 64) ? sG : (sPc - 64 * ACT_S);
      v2f bb; bb.x = src[k0 * ACT_S + vcol];
              bb.y = src[(k0 + 1) * ACT_S + vcol];
#pragma unroll
      for (int m = 0; m < 4; ++m) {
        v2f aa; aa.x = sWf[(m * 16 + lr) * 132 + k0];
                aa.y = sWf[(m * 16 + lr) * 132 + k0 + 1];
        accf[m] = wmma_f32(aa, bb, accf[m]);
      }
    }

    // GEMM3: m = Wa @ img  (attention mask logits; sigmoid>0.5 <=> logit>0)
    v8f accm[4]; accm[0]=vzero; accm[1]=vzero; accm[2]=vzero; accm[3]=vzero;
#pragma unroll
    for (int kk = 0; kk < 16; ++kk) {
      int k0 = kk * 4 + hi * 2;
      v2f bb; bb.x = sIg[k0 * ACT_S + vcol];
              bb.y = sIg[(k0 + 1) * ACT_S + vcol];
#pragma unroll
      for (int m = 0; m < 4; ++m) {
        v2f aa; aa.x = sWa[(m * 16 + lr) * 68 + k0];
                aa.y = sWa[(m * 16 + lr) * 68 + k0 + 1];
        accm[m] = wmma_f32(aa, bb, accm[m]);
      }
    }

    // epilogue: all mask algebra + store spatial_features
    const float pm = sPm[vcol], im = sIm[vcol];
    const float pmim = pm * im;
    const float t2 = pm * (1.f - im);
    const float t3 = (1.f - pm) * im;
#pragma unroll
    for (int m = 0; m < 4; ++m)
#pragma unroll
      for (int r = 0; r < 8; ++r) {
        int ch = m * 16 + r + hi * 8;
        float f   = (accf[m][r] + sBf[ch]) * pmim;
        float ifm = (b == 0) ? 1.f
                             : ((accm[m][r] + sBp[ch]) > 0.f ? 1.f : 0.f);
        float o = f + sPc[ch * ACT_S + vcol] * t2
                    + sIg[ch * ACT_S + vcol] * ifm * t3;
        out[(size_t)b * NC * NVOX + (size_t)ch * NVOX + n0 + vcol] = o;
      }
    // next iteration's wait+barrier precedes any overwrite of this tile's
    // buffers (two-deep rotation), so no trailing barrier needed
  }
}

// ---------------------------------------------------------------------------
// Kernel 4: z-reductions -> thres_map, mask, stack([mask1, mask2])
// ---------------------------------------------------------------------------
__global__ void zreduce(const float* __restrict__ wsPm,
                        const float* __restrict__ wsIm,
                        float* __restrict__ out) {
  int gid = blockIdx.x * blockDim.x + threadIdx.x;    // 32768 = B*Y*X
  int b  = gid >> 14;
  int yx = gid & (YXN - 1);
  float tmin = 1e30f, mmin = 1e30f, m1 = 0.f, m2 = 0.f;
#pragma unroll
  for (int z = 0; z < 8; ++z) {
    size_t idx = (size_t)b * NVOX + z * YXN + yx;
    float pm = wsPm[idx], im = wsIm[idx];
    float tv = pm * (1.f - im) * 0.5f + (1.f - pm) * im * 0.5f
             + (1.f - pm) * (1.f - im) * 0.5f;
    float mv = pm * im + 2.f * pm * (1.f - im) + 3.f * (1.f - pm) * im
             + 4.f * (1.f - pm) * (1.f - im);
    tmin = fminf(tmin, tv);
    mmin = fminf(mmin, mv);
    m1 = fmaxf(m1, pm);
    m2 = fmaxf(m2, im);
  }
  const size_t base = (size_t)NB * NC * NVOX;   // 16777216 spatial floats
  out[base + gid]                 = tmin;       // thres_map (2,1,128,128)
  out[base + 32768 + gid]         = mmin;       // mask      (2,1,128,128)
  out[base + 65536 + gid]         = m1;         // stack[0] = mask1
  out[base + 65536 + 32768 + gid] = m2;         // stack[1] = mask2
}

// ---------------------------------------------------------------------------
extern "C" void kernel_launch(void* const* d_in, const int* in_sizes, int n_in,
                              void* d_out, int out_size, void* d_ws,
                              size_t ws_size, hipStream_t stream) {
  const float* img = (const float*)d_in[0];
  const float* pc  = (const float*)d_in[1];
  const float* Wg  = (const float*)d_in[2];
  const float* bg  = (const float*)d_in[3];
  const float* Wf  = (const float*)d_in[4];
  const float* bf  = (const float*)d_in[5];
  const float* Wp  = (const float*)d_in[6];
  const float* bp  = (const float*)d_in[7];
  float* out = (float*)d_out;
  float* ws  = (float*)d_ws;

  float* partial = ws + WS_PARTIAL;
  float* wsWa    = ws + WS_WA;
  float* wsPm    = ws + WS_PM;
  float* wsIm    = ws + WS_IM;

  score_partials<<<NB * NSLAB, 256, K1_LDS_BYTES, stream>>>(img, pc, partial);
  softmax_wa<<<NB, 256, 0, stream>>>(partial, Wp, wsWa);
  fused_main<<<NB * 128, 256, K3_LDS_BYTES, stream>>>(
      img, pc, Wg, bg, Wf, bf, bp, wsWa, wsPm, wsIm, out);
  zreduce<<<128, 256, 0, stream>>>(wsPm, wsIm, out);
}